// GATNet_25056839205948
// MI455X (gfx1250) — compile-verified
//
#include <hip/hip_runtime.h>
#include <math.h>

// ---------------- problem constants (match reference) ----------------
#define N_NODES    25000
#define N_EDGES    400000
#define E_TOT      (N_EDGES + N_NODES)   // edges + self loops = 425000
#define NUM_GRAPHS 128
#define NEG_SLOPE  0.2f
#define H1 10
#define H2 1
#define CH 128
#define HC1 (H1 * CH)   // 1280
#define HC2 (H2 * CH)   // 128
#define F_IN 78
#define KC 32           // K-chunk staged in LDS per iteration

typedef __attribute__((ext_vector_type(2))) float v2f;
typedef __attribute__((ext_vector_type(4))) float v4f;
typedef __attribute__((ext_vector_type(8))) float v8f;

// ---------------- helpers ----------------
__device__ __forceinline__ void edge_ends(const int* __restrict__ ei, int e, int& s, int& d) {
    if (e < N_EDGES) { s = ei[e]; d = ei[N_EDGES + e]; }
    else             { s = e - N_EDGES; d = s; }          // self loop
}

// float atomic max via int/uint bitcast trick (buffer must be init'd to -inf)
__device__ __forceinline__ void atomicMaxF(float* addr, float val) {
    if (val >= 0.0f) atomicMax((int*)addr, __float_as_int(val));
    else             atomicMin((unsigned int*)addr, __float_as_uint(val));
}

// ---------------- init (vectorized b128 fill; n must be a multiple of 4) ----------------
__global__ void fill4_kernel(float* __restrict__ p, long n4, float v) {
    long i = (long)blockIdx.x * blockDim.x + threadIdx.x;
    if (i < n4) *(v4f*)(p + 4 * i) = (v4f)v;
}

// ---------------- fp32 WMMA GEMM with LDS staging ----------------
// C[M,N] = A[M,K] @ B[K,N].  N multiple of 128, K even (compile-time).
// Block = 256 threads (8 waves) computes a 16-row x 128-col C tile;
// wave w owns the 16x16 sub-tile at columns blockIdx.y*128 + w*16.
// K consumed in KC=32 LDS-staged chunks; 8 V_WMMA_F32_16X16X4_F32 per chunk.
// Hot-path loads are unconditional (clamped address + cndmask zero) so EXEC
// stays all-1s for every WMMA; chunk predicates constant-fold for full chunks.
template<int N, int K>
__global__ void __launch_bounds__(256) wmma_gemm_lds(const float* __restrict__ A,
                                                     const float* __restrict__ B,
                                                     float* __restrict__ C,
                                                     int M) {
    __shared__ float As[16 * KC];    // [row][k]   stride KC
    __shared__ float Bs[KC * 128];   // [k][col]   stride 128

    const int tid  = threadIdx.x;
    const int lane = tid & 31;
    const int wib  = tid >> 5;                 // wave in block -> n sub-tile
    const int l15  = lane & 15;
    const int kOff = (lane >> 4) << 1;         // upper half-wave holds K+2/K+3
    const int rowBase = blockIdx.x * 16;
    const int colBase = blockIdx.y * 128;
    const int bcol = wib * 16 + l15;           // column within the 128-wide tile

    // A-stage mapping: 16 rows x 16 k-pairs, one v2f per thread
    const int ar   = tid >> 4;                 // 0..15
    const int ak   = (tid & 15) << 1;          // 0,2,..,30
    const int arow = rowBase + ar;
    const size_t aRowOff = (size_t)(arow < M ? arow : M - 1) * K;
    const bool   aRowOK  = (arow < M);

    // B-stage mapping: 32 rows x 128 cols, four v4f per thread
    const int bc4 = (tid & 31) << 2;           // 0,4,..,124
    const int brb = tid >> 5;                  // 0..7

    v8f acc = {};
    for (int kc = 0; kc < K; kc += KC) {
        // ---- stage A tile (unconditional b64 load + cndmask zero) ----
        {
            const int k    = kc + ak;
            const bool ok  = aRowOK && (k < K);          // K even => pair-uniform
            const int kcl  = (k < K) ? k : 0;
            v2f av = *(const v2f*)(A + aRowOff + kcl);
            av = ok ? av : (v2f)0.0f;
            *(v2f*)&As[ar * KC + ak] = av;
        }
        // ---- stage B tile (unconditional b128 loads + cndmask zero) ----
#pragma unroll
        for (int i = 0; i < 4; ++i) {
            const int r   = brb + (i << 3);              // 0..31
            const int k   = kc + r;
            const bool ok = (k < K);
            const int kcl = ok ? k : 0;
            v4f bv = *(const v4f*)(B + (size_t)kcl * N + colBase + bc4);
            bv = ok ? bv : (v4f)0.0f;
            *(v4f*)&Bs[r * 128 + bc4] = bv;
        }
        __syncthreads();

        // ---- 8 WMMAs straight out of LDS ----
#pragma unroll
        for (int ks = 0; ks < KC / 4; ++ks) {
            const int ka = (ks << 2) + kOff;
            v2f a = *(const v2f*)&As[l15 * KC + ka];
            v2f b;
            b.x = Bs[ka * 128 + bcol];
            b.y = Bs[(ka + 1) * 128 + bcol];
            acc = __builtin_amdgcn_wmma_f32_16x16x4_f32(false, a, false, b,
                                                        (short)0, acc, false, false);
        }
        __syncthreads();
    }

    // C/D layout: VGPR r -> row r (lanes 0-15) / r+8 (lanes 16-31); lane&15 -> col
    const int rb  = rowBase + ((lane >> 4) << 3);
    const int col = colBase + bcol;
    float* Cp = C + (size_t)rb * N + col;
    if (rowBase + 16 <= M) {
        // fast path (all tiles except the last M-tile): unconditional stores,
        // compile-time immediate offsets r*N*4
#pragma unroll
        for (int r = 0; r < 8; ++r) Cp[r * N] = acc[r];
    } else {
#pragma unroll
        for (int r = 0; r < 8; ++r) {
            if (rb + r < M) Cp[r * N] = acc[r];
        }
    }
}

// ---------------- per-(node,head) attention scores: one wave per pair ----------------
__global__ void attn_scores_kernel(const float* __restrict__ h,
                                   const float* __restrict__ att_src,
                                   const float* __restrict__ att_dst,
                                   float* __restrict__ s_src, float* __restrict__ s_dst,
                                   int nH) {
    const int wid  = (int)((blockIdx.x * (long)blockDim.x + threadIdx.x) >> 5);
    const int lane = threadIdx.x & 31;
    if (wid >= N_NODES * nH) return;               // wave-uniform
    const int n  = wid / nH;
    const int hh = wid - n * nH;
    const float* hp = h + (size_t)n * nH * CH + (size_t)hh * CH;
    const float* as = att_src + (size_t)hh * CH;
    const float* ad = att_dst + (size_t)hh * CH;
    float ss = 0.0f, sd = 0.0f;
#pragma unroll
    for (int c0 = 0; c0 < CH; c0 += 32) {
        const int c = c0 + lane;
        const float v = hp[c];
        ss += v * as[c];
        sd += v * ad[c];
    }
#pragma unroll
    for (int off = 16; off > 0; off >>= 1) {
        ss += __shfl_xor(ss, off, 32);
        sd += __shfl_xor(sd, off, 32);
    }
    if (lane == 0) { s_src[wid] = ss; s_dst[wid] = sd; }
}

// ---------------- segment softmax: pass 1 (max) ----------------
__global__ void edge_max_kernel(const float* __restrict__ s_src, const float* __restrict__ s_dst,
                                const int* __restrict__ ei, float* __restrict__ e_max, int nH) {
    const int tid = blockIdx.x * blockDim.x + threadIdx.x;
    if (tid >= E_TOT * nH) return;
    const int e = tid / nH, hh = tid - e * nH;
    int s, d; edge_ends(ei, e, s, d);
    float v = s_src[s * nH + hh] + s_dst[d * nH + hh];
    v = (v > 0.0f) ? v : NEG_SLOPE * v;            // LeakyReLU(0.2)
    atomicMaxF(&e_max[d * nH + hh], v);
}

// ---------------- segment softmax: pass 2 (exp + denom) ----------------
__global__ void edge_exp_kernel(const float* __restrict__ s_src, const float* __restrict__ s_dst,
                                const int* __restrict__ ei, const float* __restrict__ e_max,
                                float* __restrict__ denom, float* __restrict__ alpha, int nH) {
    const int tid = blockIdx.x * blockDim.x + threadIdx.x;
    if (tid >= E_TOT * nH) return;
    const int e = tid / nH, hh = tid - e * nH;
    int s, d; edge_ends(ei, e, s, d);
    float v = s_src[s * nH + hh] + s_dst[d * nH + hh];
    v = (v > 0.0f) ? v : NEG_SLOPE * v;
    const float ex = __expf(v - e_max[d * nH + hh]);
    alpha[tid] = ex;                               // un-normalized for now
    atomicAdd(&denom[d * nH + hh], ex);
}

// ---------------- segment softmax: pass 3 (normalize in place) ----------------
__global__ void edge_norm_kernel(const int* __restrict__ ei, const float* __restrict__ denom,
                                 float* __restrict__ alpha, int nH) {
    const int tid = blockIdx.x * blockDim.x + threadIdx.x;
    if (tid >= E_TOT * nH) return;
    const int e = tid / nH, hh = tid - e * nH;
    int s, d; edge_ends(ei, e, s, d);
    alpha[tid] = alpha[tid] / denom[d * nH + hh];
}

// ---------------- edge aggregation: out[dst] += h[src] * alpha ----------------
// One block per edge. HC multiple of 4*blockDim not required; HC%4==0 assumed.
// Vectorized b128 gather; head index of element i is i>>7 (C=128).
__global__ void aggregate_kernel(const float* __restrict__ h, const float* __restrict__ alpha,
                                 const int* __restrict__ ei, float* __restrict__ out, int nH) {
    const int e = blockIdx.x;
    int s, d; edge_ends(ei, e, s, d);
    const int HC = nH * CH;
    const float* hs = h + (size_t)s * HC;
    float* od = out + (size_t)d * HC;
    const float* al = alpha + (size_t)e * nH;
    for (int i4 = threadIdx.x; i4 * 4 < HC; i4 += blockDim.x) {
        const int i = i4 * 4;                     // 4 consecutive channels, same head
        const float a = al[i >> 7];
        const v4f hv = *(const v4f*)(hs + i);
        atomicAdd(&od[i + 0], hv.x * a);
        atomicAdd(&od[i + 1], hv.y * a);
        atomicAdd(&od[i + 2], hv.z * a);
        atomicAdd(&od[i + 3], hv.w * a);
    }
}

// ---------------- bias add + ELU ----------------
__global__ void bias_elu_kernel(float* __restrict__ x, const float* __restrict__ bias,
                                int HC, long total) {
    const long i = (long)blockIdx.x * blockDim.x + threadIdx.x;
    if (i >= total) return;
    float v = x[i] + bias[i % HC];
    x[i] = (v > 0.0f) ? v : (__expf(v) - 1.0f);    // ELU(alpha=1)
}

// ---------------- global max pool ----------------
__global__ void pool_kernel(const float* __restrict__ h, const int* __restrict__ batch,
                            float* __restrict__ pooled) {
    const int tid = blockIdx.x * blockDim.x + threadIdx.x;
    if (tid >= N_NODES * CH) return;
    const int n = tid >> 7, c = tid & 127;
    const int g = batch[n];
    atomicMaxF(&pooled[(size_t)g * CH + c], h[tid]);
}

// =====================================================================
extern "C" void kernel_launch(void* const* d_in, const int* in_sizes, int n_in,
                              void* d_out, int out_size, void* d_ws, size_t ws_size,
                              hipStream_t stream) {
    const float* x1       = (const float*)d_in[0];
    const float* W1       = (const float*)d_in[1];
    const float* att_src1 = (const float*)d_in[2];
    const float* att_dst1 = (const float*)d_in[3];
    const float* bias1    = (const float*)d_in[4];
    const float* W2       = (const float*)d_in[5];
    const float* att_src2 = (const float*)d_in[6];
    const float* att_dst2 = (const float*)d_in[7];
    const float* bias2    = (const float*)d_in[8];
    const int*   ei       = (const int*)d_in[9];
    const int*   batch    = (const int*)d_in[10];

    float* pooled = (float*)d_out;                         // [128,128]
    float* alpha1 = pooled + (size_t)NUM_GRAPHS * CH;      // [425000,10]

    // ---- workspace layout (floats) ----
    float* ws    = (float*)d_ws;
    float* h1    = ws;                 ws += (size_t)N_NODES * HC1;   // 32.0M
    float* out1  = ws;                 ws += (size_t)N_NODES * HC1;   // 32.0M
    float* ssrc1 = ws;                 ws += (size_t)N_NODES * H1;
    float* sdst1 = ws;                 ws += (size_t)N_NODES * H1;
    float* emax1 = ws;                 ws += (size_t)N_NODES * H1;
    float* den1  = ws;                 ws += (size_t)N_NODES * H1;
    float* h2    = ws;                 ws += (size_t)N_NODES * HC2;
    float* out2  = ws;                 ws += (size_t)N_NODES * HC2;
    float* ssrc2 = ws;                 ws += (size_t)N_NODES * H2;
    float* sdst2 = ws;                 ws += (size_t)N_NODES * H2;
    float* emax2 = ws;                 ws += (size_t)N_NODES * H2;
    float* den2  = ws;                 ws += (size_t)N_NODES * H2;
    float* alpha2= ws;                 ws += (size_t)E_TOT;

    const float NEG_INF = -INFINITY;
    const int TB = 256;
    auto blocks  = [](long n, int tb) { return (unsigned)((n + tb - 1) / tb); };
    auto fblocks = [](long n, int tb) { return (unsigned)((n / 4 + tb - 1) / tb); };

    // ---- init accumulators (all sizes are multiples of 4) ----
    fill4_kernel<<<fblocks((long)N_NODES * HC1, TB), TB, 0, stream>>>(out1, (long)N_NODES * HC1 / 4, 0.0f);
    fill4_kernel<<<fblocks((long)N_NODES * H1,  TB), TB, 0, stream>>>(den1, (long)N_NODES * H1 / 4, 0.0f);
    fill4_kernel<<<fblocks((long)N_NODES * H1,  TB), TB, 0, stream>>>(emax1, (long)N_NODES * H1 / 4, NEG_INF);
    fill4_kernel<<<fblocks((long)N_NODES * HC2, TB), TB, 0, stream>>>(out2, (long)N_NODES * HC2 / 4, 0.0f);
    fill4_kernel<<<fblocks((long)N_NODES * H2,  TB), TB, 0, stream>>>(den2, (long)N_NODES * H2 / 4, 0.0f);
    fill4_kernel<<<fblocks((long)N_NODES * H2,  TB), TB, 0, stream>>>(emax2, (long)N_NODES * H2 / 4, NEG_INF);
    fill4_kernel<<<fblocks((long)NUM_GRAPHS * CH, TB), TB, 0, stream>>>(pooled, (long)NUM_GRAPHS * CH / 4, NEG_INF);

    const int mTiles = (N_NODES + 15) / 16;                // 1563

    // ================= Layer 1 =================
    {   // GEMM1: h1 = x1 @ W1   (M=25000, N=1280, K=78) — fp32 WMMA, LDS-staged
        dim3 grid(mTiles, HC1 / 128);                      // (1563, 10)
        wmma_gemm_lds<HC1, F_IN><<<grid, 256, 0, stream>>>(x1, W1, h1, N_NODES);
    }
    attn_scores_kernel<<<blocks((long)N_NODES * H1 * 32, TB), TB, 0, stream>>>(
        h1, att_src1, att_dst1, ssrc1, sdst1, H1);
    edge_max_kernel<<<blocks((long)E_TOT * H1, TB), TB, 0, stream>>>(ssrc1, sdst1, ei, emax1, H1);
    edge_exp_kernel<<<blocks((long)E_TOT * H1, TB), TB, 0, stream>>>(ssrc1, sdst1, ei, emax1, den1, alpha1, H1);
    edge_norm_kernel<<<blocks((long)E_TOT * H1, TB), TB, 0, stream>>>(ei, den1, alpha1, H1);
    aggregate_kernel<<<E_TOT, 256, 0, stream>>>(h1, alpha1, ei, out1, H1);
    bias_elu_kernel<<<blocks((long)N_NODES * HC1, TB), TB, 0, stream>>>(
        out1, bias1, HC1, (long)N_NODES * HC1);

    // ================= Layer 2 =================
    {   // GEMM2: h2 = out1 @ W2  (M=25000, N=128, K=1280) — fp32 WMMA, LDS-staged
        dim3 grid(mTiles, HC2 / 128);                      // (1563, 1)
        wmma_gemm_lds<HC2, HC1><<<grid, 256, 0, stream>>>(out1, W2, h2, N_NODES);
    }
    attn_scores_kernel<<<blocks((long)N_NODES * H2 * 32, TB), TB, 0, stream>>>(
        h2, att_src2, att_dst2, ssrc2, sdst2, H2);
    edge_max_kernel<<<blocks((long)E_TOT * H2, TB), TB, 0, stream>>>(ssrc2, sdst2, ei, emax2, H2);
    edge_exp_kernel<<<blocks((long)E_TOT * H2, TB), TB, 0, stream>>>(ssrc2, sdst2, ei, emax2, den2, alpha2, H2);
    edge_norm_kernel<<<blocks((long)E_TOT * H2, TB), TB, 0, stream>>>(ei, den2, alpha2, H2);
    aggregate_kernel<<<E_TOT, 128, 0, stream>>>(h2, alpha2, ei, out2, H2);
    bias_elu_kernel<<<blocks((long)N_NODES * HC2, TB), TB, 0, stream>>>(
        out2, bias2, HC2, (long)N_NODES * HC2);

    // ================= Pool =================
    pool_kernel<<<blocks((long)N_NODES * CH, TB), TB, 0, stream>>>(out2, batch, pooled);
}